// localStack_layer_51213190038252
// MI455X (gfx1250) — compile-verified
//
#include <hip/hip_runtime.h>
#include <cstdint>
#include <cstddef>

// Problem shape (fixed by the reference).
static constexpr int Bn = 8;
static constexpr int Cn = 3;
static constexpr int Hn = 512;
static constexpr int Wn = 512;
static constexpr int TH = 4;          // output rows per block
static constexpr int HALO = TH + 4;   // input rows staged (j in [-2,2])
static constexpr int THREADS = 256;   // 8 wave32 waves

typedef float v4f __attribute__((ext_vector_type(4)));

// out[b, s*3+c, h, w] = img[b, c, (h-j)&511, (w-i)&511],  s=(i+2)*5+(j+2)
__global__ __launch_bounds__(THREADS)
void roll_stack_kernel(const float* __restrict__ img, float* __restrict__ out) {
    __shared__ __align__(16) float smem[HALO * Wn];   // 16 KB

    const int t    = threadIdx.x;
    const int hblk = blockIdx.x & (Hn / TH - 1);      // 128 h-tiles
    const int bc   = blockIdx.x >> 7;
    const int c    = bc % Cn;
    const int b    = bc / Cn;
    const int h0   = hblk * TH;

    // LDS byte address = low 32 bits of the generic (flat) shared address.
    const uint32_t lds_base = (uint32_t)(uintptr_t)(&smem[0]);

    // ---- Phase 1: async DMA 8 input rows (with +/-2 halo) into LDS ----
    // HALO*Wn floats = 1024 float4 chunks; 4 per thread.
    const float* src_base = img + (size_t)(b * Cn + c) * (Hn * Wn);
#pragma unroll
    for (int q = 0; q < (HALO * Wn) / (4 * THREADS); ++q) {
        const int l    = q * THREADS + t;     // float4 id within tile
        const int row  = l >> 7;              // Wn/4 = 128 float4 per row
        const int col4 = l & 127;
        const int gh   = (h0 - 2 + row) & (Hn - 1);
        const uint64_t gp =
            (uint64_t)(uintptr_t)(src_base + (size_t)gh * Wn + col4 * 4);
        const uint32_t lp = lds_base + (uint32_t)((row * Wn + col4 * 4) * 4);
        asm volatile("global_load_async_to_lds_b128 %0, %1, off"
                     :: "v"(lp), "v"(gp)
                     : "memory");
    }
    asm volatile("s_wait_asynccnt 0x0" ::: "memory");
    __syncthreads();

    // ---- Phase 2: for each H-shift row, load one 8-float LDS window and
    //      emit all 5 W-shifted float4s as non-temporal b128 stores ----
    float* dstb = out + ((size_t)b * (25 * Cn) + c) * (size_t)(Hn * Wn);

    for (int jj = 0; jj < 5; ++jj) {          // j = jj - 2 (H shift)
#pragma unroll
        for (int k = 0; k < 2; ++k) {         // 256 threads x 2 = TH*128 float4
            const int idx2 = k * THREADS + t; // 0..511
            const int row  = idx2 >> 7;       // 0..3
            const int col4 = idx2 & 127;
            const int lrow = row + 4 - jj;    // 0..7 (inside halo)
            const float* srow = &smem[lrow * Wn];
            const int w0 = col4 * 4;

            // Window srow[(w0+d)&511], d = -2..5 : aligned b128 mid + 4 halo b32
            float wnd[8];
            const v4f mid = *reinterpret_cast<const v4f*>(&srow[w0]);
            wnd[0] = srow[(w0 - 2) & (Wn - 1)];
            wnd[1] = srow[(w0 - 1) & (Wn - 1)];
            wnd[2] = mid.x; wnd[3] = mid.y; wnd[4] = mid.z; wnd[5] = mid.w;
            wnd[6] = srow[(w0 + 4) & (Wn - 1)];
            wnd[7] = srow[(w0 + 5) & (Wn - 1)];

            float* drow = dstb + (size_t)(h0 + row) * Wn + w0;
#pragma unroll
            for (int ii = 0; ii < 5; ++ii) {  // i = ii - 2 (W shift)
                const int s = ii * 5 + jj;    // shift index in reference order
                // v_k = srow[(w0 + k - i)&511] = wnd[k + 4 - ii]
                v4f v;
                v.x = wnd[4 - ii + 0];
                v.y = wnd[4 - ii + 1];
                v.z = wnd[4 - ii + 2];
                v.w = wnd[4 - ii + 3];
                v4f* dp = reinterpret_cast<v4f*>(drow + (size_t)(s * Cn) * (Hn * Wn));
                __builtin_nontemporal_store(v, dp);
            }
        }
    }
}

extern "C" void kernel_launch(void* const* d_in, const int* in_sizes, int n_in,
                              void* d_out, int out_size, void* d_ws, size_t ws_size,
                              hipStream_t stream) {
    (void)in_sizes; (void)n_in; (void)out_size; (void)d_ws; (void)ws_size;
    const float* img = (const float*)d_in[0];
    float* out = (float*)d_out;
    const int grid = Bn * Cn * (Hn / TH);   // 3072 blocks
    roll_stack_kernel<<<grid, THREADS, 0, stream>>>(img, out);
}